// AdditiveIntervention_39084202394056
// MI455X (gfx1250) — compile-verified
//
#include <hip/hip_runtime.h>

typedef __attribute__((ext_vector_type(2))) float v2f;
typedef __attribute__((ext_vector_type(8))) float v8f;

#define B_ 512
#define N_ 1024
#define D_ 1024
#define FUSE_ 1024
#define A_ 256

__device__ __forceinline__ float tanh_hw(float x) {
    float r;
    asm("v_tanh_f32 %0, %1" : "=v"(r) : "v"(x));
    return r;
}

// ---------------------------------------------------------------------------
// Kernel 1: combined QK projection GEMM.
//   rows 0..511    : Q = fuse_rep[512,1024] @ Wq[1024,256]
//   rows 512..1535 : K = confounder[1024,1024] @ Wk[1024,256]
// One wave per 16x16 output tile, V_WMMA_F32_16X16X4_F32, K-loop step 4.
// ---------------------------------------------------------------------------
__global__ __launch_bounds__(256) void qk_gemm_kernel(
    const float* __restrict__ conf, const float* __restrict__ fuse,
    const float* __restrict__ Wq, const float* __restrict__ Wk,
    float* __restrict__ qk) {
    const int lane = threadIdx.x & 31;
    const int wave = threadIdx.x >> 5;
    const int tile = blockIdx.x * 8 + wave;      // 1536 tiles total
    const int tilesPerRow = A_ / 16;             // 16
    const int tm = tile / tilesPerRow;           // 0..95 (row tile)
    const int tn = tile % tilesPerRow;           // 0..15 (col tile)
    const int half = lane >> 4;                  // 0 or 1
    const int l15 = lane & 15;

    // A operand: row = M (lane&15), K covered 2 per VGPR per half
    const float* Arow;
    const float* W;
    if (tm < B_ / 16) {                          // Q part (wave-uniform branch)
        Arow = fuse + (tm * 16 + l15) * FUSE_;
        W = Wq;
    } else {                                     // K part
        Arow = conf + ((tm - B_ / 16) * 16 + l15) * D_;
        W = Wk;
    }
    const int col = tn * 16 + l15;               // B/C/D column per lane

    v8f acc = {0.f, 0.f, 0.f, 0.f, 0.f, 0.f, 0.f, 0.f};
    for (int k = 0; k < FUSE_; k += 4) {
        const int kb = k + 2 * half;
        v2f a = *(const v2f*)(Arow + kb);        // A[m][kb], A[m][kb+1]
        v2f b;
        b.x = W[(kb + 0) * A_ + col];            // B[kb][col]
        b.y = W[(kb + 1) * A_ + col];            // B[kb+1][col]
        acc = __builtin_amdgcn_wmma_f32_16x16x4_f32(
            false, a, false, b, (short)0, acc, false, false);
    }
#pragma unroll
    for (int r = 0; r < 8; ++r) {                // D: VGPR r -> M = r + 8*half
        qk[(tm * 16 + r + 8 * half) * A_ + col] = acc[r];
    }
}

// ---------------------------------------------------------------------------
// Kernel 2: fused additive-attention score (never materializes [B,N,A]).
//   scores[b,n] = sum_a wt[a] * tanh(q[b,a] + k[n,a])
// Grid: (N/256, B). q-row and wt staged in LDS; k-row streamed float4/lane.
// ---------------------------------------------------------------------------
__global__ __launch_bounds__(256) void scores_kernel(
    const float* __restrict__ qk, const float* __restrict__ wt,
    float* __restrict__ scores) {
    __shared__ float qs[A_];
    __shared__ float wv[A_];
    const int tid = threadIdx.x;
    const int b = blockIdx.y;
    const int n = blockIdx.x * 256 + tid;

    qs[tid] = qk[b * A_ + tid];                  // q row for this b
    wv[tid] = wt[tid];
    __syncthreads();

    const float* krow = qk + (B_ + n) * A_;      // k row for this n
    float s = 0.f;
#pragma unroll 4
    for (int a = 0; a < A_; a += 4) {
        float4 kv = *(const float4*)(krow + a);
        s += wv[a + 0] * tanh_hw(qs[a + 0] + kv.x);
        s += wv[a + 1] * tanh_hw(qs[a + 1] + kv.y);
        s += wv[a + 2] * tanh_hw(qs[a + 2] + kv.z);
        s += wv[a + 3] * tanh_hw(qs[a + 3] + kv.w);
    }
    scores[b * N_ + n] = s;
}

// ---------------------------------------------------------------------------
// Kernel 3: in-place softmax over N, scaled by probabilities[n].
// One block per b-row; 256 threads handle 4 values each.
// ---------------------------------------------------------------------------
__global__ __launch_bounds__(256) void softmax_kernel(
    float* __restrict__ scores, const float* __restrict__ probs) {
    __shared__ float red[256];
    const int b = blockIdx.x;
    const int tid = threadIdx.x;

    float v[4];
    float mx = -3.402823466e38f;
#pragma unroll
    for (int i = 0; i < 4; ++i) {
        v[i] = scores[b * N_ + tid + i * 256];
        mx = fmaxf(mx, v[i]);
    }
    red[tid] = mx;
    __syncthreads();
    for (int s = 128; s > 0; s >>= 1) {
        if (tid < s) red[tid] = fmaxf(red[tid], red[tid + s]);
        __syncthreads();
    }
    mx = red[0];
    __syncthreads();

    float sum = 0.f;
#pragma unroll
    for (int i = 0; i < 4; ++i) {
        v[i] = __expf(v[i] - mx);
        sum += v[i];
    }
    red[tid] = sum;
    __syncthreads();
    for (int s = 128; s > 0; s >>= 1) {
        if (tid < s) red[tid] += red[tid + s];
        __syncthreads();
    }
    const float inv = 1.f / red[0];
#pragma unroll
    for (int i = 0; i < 4; ++i) {
        const int n = tid + i * 256;
        scores[b * N_ + n] = v[i] * inv * probs[n];
    }
}

// ---------------------------------------------------------------------------
// Kernel 4: fin = wmat[512,1024] @ confounder[1024,1024] -> d_out [512,1024].
// Same f32 WMMA tiling as kernel 1.
// ---------------------------------------------------------------------------
__global__ __launch_bounds__(256) void fin_gemm_kernel(
    const float* __restrict__ wmat, const float* __restrict__ conf,
    float* __restrict__ out) {
    const int lane = threadIdx.x & 31;
    const int wave = threadIdx.x >> 5;
    const int tile = blockIdx.x * 8 + wave;      // 2048 tiles total
    const int tilesPerRow = D_ / 16;             // 64
    const int tm = tile / tilesPerRow;           // 0..31
    const int tn = tile % tilesPerRow;           // 0..63
    const int half = lane >> 4;
    const int l15 = lane & 15;

    const float* Arow = wmat + (tm * 16 + l15) * N_;
    const int col = tn * 16 + l15;

    v8f acc = {0.f, 0.f, 0.f, 0.f, 0.f, 0.f, 0.f, 0.f};
    for (int k = 0; k < N_; k += 4) {
        const int kb = k + 2 * half;
        v2f a = *(const v2f*)(Arow + kb);
        v2f b;
        b.x = conf[(kb + 0) * D_ + col];
        b.y = conf[(kb + 1) * D_ + col];
        acc = __builtin_amdgcn_wmma_f32_16x16x4_f32(
            false, a, false, b, (short)0, acc, false, false);
    }
#pragma unroll
    for (int r = 0; r < 8; ++r) {
        out[(tm * 16 + r + 8 * half) * D_ + col] = acc[r];
    }
}

// ---------------------------------------------------------------------------
extern "C" void kernel_launch(void* const* d_in, const int* in_sizes, int n_in,
                              void* d_out, int out_size, void* d_ws, size_t ws_size,
                              hipStream_t stream) {
    const float* conf  = (const float*)d_in[0];  // [N, D]
    const float* fuse  = (const float*)d_in[1];  // [B, FUSE]
    const float* probs = (const float*)d_in[2];  // [1, N]
    const float* Wq    = (const float*)d_in[3];  // [FUSE, A]
    const float* Wk    = (const float*)d_in[4];  // [D, A]
    const float* wt    = (const float*)d_in[5];  // [A]
    float* out = (float*)d_out;                  // [B, D]

    float* qk     = (float*)d_ws;                // (B+N) x A = 1536*256 floats
    float* scores = qk + (B_ + N_) * A_;         // B x N    =  512*1024 floats

    // 1536 tiles / 8 waves per block = 192 blocks
    qk_gemm_kernel<<<dim3(192), dim3(256), 0, stream>>>(conf, fuse, Wq, Wk, qk);
    // fused tanh scoring: grid (N/256, B)
    scores_kernel<<<dim3(N_ / 256, B_), dim3(256), 0, stream>>>(qk, wt, scores);
    // softmax * probs, in place
    softmax_kernel<<<dim3(B_), dim3(256), 0, stream>>>(scores, probs);
    // 2048 tiles / 8 waves per block = 256 blocks
    fin_gemm_kernel<<<dim3(256), dim3(256), 0, stream>>>(scores, conf, out);
}